// PSTOp_29600914604801
// MI455X (gfx1250) — compile-verified
//
#include <hip/hip_runtime.h>

#define B_  4
#define L_  8
#define N_  4096
#define C_  3
#define M_  2048
#define K_  32
#define R2_ 0.25f   // RADIUS^2

typedef __attribute__((ext_vector_type(2))) float v2f;
typedef __attribute__((ext_vector_type(8))) float v8f;

// D = A(16x4 f32) * B(4x16 f32) + C(16x16 f32)
__device__ __forceinline__ v8f wmma_f32(v2f a, v2f b, v8f c) {
    return __builtin_amdgcn_wmma_f32_16x16x4_f32(
        /*neg_a=*/false, a, /*neg_b=*/false, b,
        /*c_mod=*/(short)0, c, /*reuse_a=*/false, /*reuse_b=*/false);
}

// Copy one frame (N_*3 floats = 48KB) from global to LDS using the CDNA5
// async-to-LDS path (ASYNCcnt). 256 threads x 16B per iteration.
// Caller must __syncthreads() afterwards; each wave waits its own ASYNCcnt here.
__device__ __forceinline__ void async_frame_to_lds(float* dst_lds,
                                                   const float* __restrict__ src,
                                                   int tid) {
    const unsigned lds0 =
        (unsigned)(uintptr_t)(__attribute__((address_space(3))) float*)dst_lds;
#pragma unroll
    for (int it = 0; it < (N_ * 3 * 4) / (256 * 16); ++it) {   // 12 iterations
        const unsigned byte_off = (unsigned)(it * 4096 + tid * 16);
        asm volatile("global_load_async_to_lds_b128 %0, %1, off"
                     :: "v"(lds0 + byte_off),
                        "v"((const void*)((const char*)src + byte_off))
                     : "memory");
    }
    asm volatile("s_wait_asynccnt 0x0" ::: "memory");
}

// ---------------------------------------------------------------------------
// Kernel 1: furthest point sampling, one block per (b, t). Writes anchor xyz
// straight into the d_out xyz region [B,8,M,3].
// ---------------------------------------------------------------------------
__global__ __launch_bounds__(256) void fps_kernel(const float* __restrict__ xyzs,
                                                  float* __restrict__ out_xyz) {
    const int bt = blockIdx.x;                       // b*8 + t  (frame f == t)
    const float* fx = xyzs + (size_t)bt * N_ * 3;
    float* oxyz = out_xyz + (size_t)bt * M_ * 3;

    __shared__ float sxyz[N_ * 3];
    __shared__ float pv[8];
    __shared__ int   pi[8];
    __shared__ int   ssel;

    const int tid = threadIdx.x;
    async_frame_to_lds(sxyz, fx, tid);
    __syncthreads();

    // each thread owns 16 contiguous points -> argmax tie-break = lowest index
    float px[16], py[16], pz[16], dist[16];
#pragma unroll
    for (int u = 0; u < 16; ++u) {
        const int n = tid * 16 + u;
        px[u] = sxyz[n * 3 + 0];
        py[u] = sxyz[n * 3 + 1];
        pz[u] = sxyz[n * 3 + 2];
        dist[u] = 1e10f;
    }

    const int lane = tid & 31;
    const int wave = tid >> 5;
    int last = 0;

    for (int m = 0; m < M_; ++m) {
        if (tid == 0) {
            oxyz[m * 3 + 0] = sxyz[last * 3 + 0];
            oxyz[m * 3 + 1] = sxyz[last * 3 + 1];
            oxyz[m * 3 + 2] = sxyz[last * 3 + 2];
        }
        const float lx = sxyz[last * 3 + 0];
        const float ly = sxyz[last * 3 + 1];
        const float lz = sxyz[last * 3 + 2];

        float bestv = -1.0f;
        int   besti = 0x7fffffff;
#pragma unroll
        for (int u = 0; u < 16; ++u) {
            const float dx = px[u] - lx, dy = py[u] - ly, dz = pz[u] - lz;
            const float d  = dx * dx + dy * dy + dz * dz;
            const float nd = fminf(dist[u], d);
            dist[u] = nd;
            const int n = tid * 16 + u;
            if (nd > bestv || (nd == bestv && n < besti)) { bestv = nd; besti = n; }
        }
        // wave32 reduction
#pragma unroll
        for (int off = 16; off >= 1; off >>= 1) {
            const float ov = __shfl_xor(bestv, off, 32);
            const int   oi = __shfl_xor(besti, off, 32);
            if (ov > bestv || (ov == bestv && oi < besti)) { bestv = ov; besti = oi; }
        }
        if (lane == 0) { pv[wave] = bestv; pi[wave] = besti; }
        __syncthreads();
        if (tid == 0) {
            float bv = pv[0]; int bi = pi[0];
            for (int w = 1; w < 8; ++w)
                if (pv[w] > bv || (pv[w] == bv && pi[w] < bi)) { bv = pv[w]; bi = pi[w]; }
            ssel = bi;
        }
        __syncthreads();
        last = ssel;
    }
}

// ---------------------------------------------------------------------------
// Kernel 2: fused ball-query + gather + MLP (WMMA f32) + k-max + temporal mix.
// One block = 16 anchors of one (b, t). 8 waves; each wave serves 2 anchors.
// ---------------------------------------------------------------------------
__global__ __launch_bounds__(256) void pst_kernel(const float* __restrict__ xyzs,
                                                  const float* __restrict__ feats,
                                                  const float* __restrict__ Ws0,   // [64,6]
                                                  const float* __restrict__ Ws1,   // [128,64]
                                                  const float* __restrict__ Wt,    // [3,256,128]
                                                  const float* __restrict__ anchors, // [B,8,M,3]
                                                  float* __restrict__ out_feat) {  // [B,8,256,M]
    __shared__ float sxyz[N_ * 3];       // 48 KB  neighbor frame
    __shared__ float sf[8][8][K_];       //  8 KB  per-wave SF (K padded 6->8)
    __shared__ float h1[8][64][K_];      // 64 KB  per-wave layer-1 activations
    __shared__ float hmax[16][132];      // ~8 KB  per-anchor 128-ch max (padded stride)

    const int bid   = blockIdx.x;
    const int mt    = bid & 127;
    const int t     = (bid >> 7) & 7;
    const int b     = bid >> 10;
    const int mbase = mt * 16;

    const int tid  = threadIdx.x;
    const int lane = tid & 31;
    const int wave = tid >> 5;
    const int half = lane >> 4;          // K split across lane halves
    const int l16  = lane & 15;

    v8f st0 = {};                        // out channels [wave*32 .. wave*32+15]
    v8f st1 = {};                        // out channels [wave*32+16 .. +31]

    for (int j = 0; j < 3; ++j) {
        int f = t + j - 1;
        f = f < 0 ? 0 : (f > 7 ? 7 : f); // temporal replicate padding
        const float* fx = xyzs  + (size_t)(b * L_ + f) * N_ * 3;
        const float* ff = feats + (size_t)(b * L_ + f) * C_ * N_;

        __syncthreads();                                     // protect sxyz / hmax reuse
        async_frame_to_lds(sxyz, fx, tid);                   // CDNA5 async -> LDS
        __syncthreads();

        for (int sub = 0; sub < 2; ++sub) {
            const int alocal = wave * 2 + sub;
            const int m      = mbase + alocal;
            const size_t abase = ((size_t)(b * L_ + t) * M_ + m) * 3;
            const float ax = anchors[abase + 0];
            const float ay = anchors[abase + 1];
            const float az = anchors[abase + 2];

            // ---- ball query: first 32 in-radius indices (ascending) ----
            int cnt = 0, myidx = -1;
            for (int base = 0; base < N_ && cnt < K_; base += 32) {
                const int n = base + lane;
                const float dx = sxyz[n * 3 + 0] - ax;
                const float dy = sxyz[n * 3 + 1] - ay;
                const float dz = sxyz[n * 3 + 2] - az;
                unsigned msk = (unsigned)__ballot(dx * dx + dy * dy + dz * dz < R2_);
                while (msk) {
                    const int bit = __ffs(msk) - 1;
                    msk &= msk - 1;
                    if (cnt == lane) myidx = base + bit;
                    if (++cnt >= K_) break;
                }
            }
            int first = __shfl(myidx, 0, 32);
            if (first < 0) first = 0;
            if (myidx < 0) myidx = first;

            // ---- gather: lane == sample k ----
            {
                const int p = myidx;
                sf[wave][0][lane] = sxyz[p * 3 + 0] - ax;
                sf[wave][1][lane] = sxyz[p * 3 + 1] - ay;
                sf[wave][2][lane] = sxyz[p * 3 + 2] - az;
                sf[wave][3][lane] = ff[0 * N_ + p];
                sf[wave][4][lane] = ff[1 * N_ + p];
                sf[wave][5][lane] = ff[2 * N_ + p];
                sf[wave][6][lane] = 0.0f;
                sf[wave][7][lane] = 0.0f;
            }

            // ---- layer 1: relu(Ws0[64x6] @ SF[6x32]) via WMMA f32 16x16x4 ----
            for (int rt = 0; rt < 4; ++rt) {
                v8f c0 = {}, c1 = {};
#pragma unroll
                for (int kk = 0; kk < 2; ++kk) {
                    const int k0  = 4 * kk + 2 * half;     // A layout: K 0/2 in halves
                    const int row = rt * 16 + l16;
                    v2f a;
                    a.x = (k0     < 6) ? Ws0[row * 6 + k0]     : 0.0f;
                    a.y = (k0 + 1 < 6) ? Ws0[row * 6 + k0 + 1] : 0.0f;
                    v2f b0, b1;
                    b0.x = sf[wave][k0][l16];      b0.y = sf[wave][k0 + 1][l16];
                    b1.x = sf[wave][k0][16 + l16]; b1.y = sf[wave][k0 + 1][16 + l16];
                    c0 = wmma_f32(a, b0, c0);
                    c1 = wmma_f32(a, b1, c1);
                }
#pragma unroll
                for (int p = 0; p < 8; ++p) {              // D: lanes=N, VGPR p -> M=p(+8)
                    const int row = rt * 16 + p + 8 * half;
                    h1[wave][row][l16]      = fmaxf(c0[p], 0.0f);
                    h1[wave][row][16 + l16] = fmaxf(c1[p], 0.0f);
                }
            }

            // ---- layer 2: relu(Ws1[128x64] @ H1[64x32]), then max over k ----
            for (int rt = 0; rt < 8; ++rt) {
                v8f c0 = {}, c1 = {};
                for (int ks = 0; ks < 16; ++ks) {
                    const int k0  = 4 * ks + 2 * half;
                    const int row = rt * 16 + l16;
                    v2f a;
                    a.x = Ws1[row * 64 + k0];
                    a.y = Ws1[row * 64 + k0 + 1];
                    v2f b0, b1;
                    b0.x = h1[wave][k0][l16];      b0.y = h1[wave][k0 + 1][l16];
                    b1.x = h1[wave][k0][16 + l16]; b1.y = h1[wave][k0 + 1][16 + l16];
                    c0 = wmma_f32(a, b0, c0);
                    c1 = wmma_f32(a, b1, c1);
                }
#pragma unroll
                for (int p = 0; p < 8; ++p) {
                    float v = fmaxf(fmaxf(c0[p], 0.0f), fmaxf(c1[p], 0.0f));
#pragma unroll
                    for (int off = 1; off < 16; off <<= 1)  // stays within 16-lane half
                        v = fmaxf(v, __shfl_xor(v, off, 32));
                    if (l16 == 0) hmax[alocal][rt * 16 + p + 8 * half] = v;
                }
            }
        } // sub

        __syncthreads();  // hmax for all 16 anchors ready

        // ---- temporal mix: st += relu(Wt[j][256x128] @ Hmax[128x16]) ----
#pragma unroll
        for (int tt = 0; tt < 2; ++tt) {
            const int rt = wave * 2 + tt;
            v8f c = {};
            for (int ks = 0; ks < 32; ++ks) {
                const int k0  = 4 * ks + 2 * half;
                const int row = rt * 16 + l16;
                v2f a;
                a.x = Wt[((size_t)j * 256 + row) * 128 + k0];
                a.y = Wt[((size_t)j * 256 + row) * 128 + k0 + 1];
                v2f bb;
                bb.x = hmax[l16][k0];
                bb.y = hmax[l16][k0 + 1];
                c = wmma_f32(a, bb, c);
            }
#pragma unroll
            for (int p = 0; p < 8; ++p) {
                const float r = fmaxf(c[p], 0.0f);
                if (tt == 0) st0[p] += r; else st1[p] += r;
            }
        }
    } // j

    // ---- store [B,8,256,M] ----
#pragma unroll
    for (int p = 0; p < 8; ++p) {
        const int oc0 = (wave * 2 + 0) * 16 + p + 8 * half;
        const int oc1 = (wave * 2 + 1) * 16 + p + 8 * half;
        const int m   = mbase + l16;
        out_feat[((size_t)(b * L_ + t) * 256 + oc0) * M_ + m] = st0[p];
        out_feat[((size_t)(b * L_ + t) * 256 + oc1) * M_ + m] = st1[p];
    }
}

// ---------------------------------------------------------------------------
extern "C" void kernel_launch(void* const* d_in, const int* in_sizes, int n_in,
                              void* d_out, int out_size, void* d_ws, size_t ws_size,
                              hipStream_t stream) {
    (void)in_sizes; (void)n_in; (void)out_size; (void)d_ws; (void)ws_size;
    const float* xyzs  = (const float*)d_in[0];  // [4,8,4096,3]
    const float* feats = (const float*)d_in[1];  // [4,8,3,4096]
    const float* Ws0   = (const float*)d_in[2];  // [64,6]
    const float* Ws1   = (const float*)d_in[3];  // [128,64]
    const float* Wt    = (const float*)d_in[4];  // [3,256,128]

    float* out      = (float*)d_out;
    float* out_xyz  = out;                                   // [4,8,2048,3]
    float* out_feat = out + (size_t)B_ * L_ * M_ * 3;        // [4,8,256,2048]

    fps_kernel<<<B_ * L_, 256, 0, stream>>>(xyzs, out_xyz);
    pst_kernel<<<B_ * L_ * (M_ / 16), 256, 0, stream>>>(xyzs, feats, Ws0, Ws1, Wt,
                                                        out_xyz, out_feat);
}